// Encoder_Base_44813688767276
// MI455X (gfx1250) — compile-verified
//
#include <hip/hip_runtime.h>
#include <hip/hip_bf16.h>
#include <math.h>

typedef __attribute__((ext_vector_type(2))) float v2f;
typedef __attribute__((ext_vector_type(8))) float v8f;

#define TV_TAU   0.25f
#define TV_RATIO 2.5f   /* tau / weight = 0.25 / 0.1 */

__device__ __forceinline__ float silu_f(float v) {
    return v * (1.0f / (1.0f + __expf(-v)));
}

// ---------------------------------------------------------------- zero fill
__global__ void k_zero(float* __restrict__ p, int n) {
    int i = blockIdx.x * blockDim.x + threadIdx.x;
    if (i < n) p[i] = 0.0f;
}

// ---------------------------------------------------------------- TV Chambolle
// p layout: [img][2][H][W].  out(i,j) = img + div(p) recomputed locally.
__device__ __forceinline__ float tv_outval(const float* __restrict__ im,
                                           const float* __restrict__ p0,
                                           const float* __restrict__ p1,
                                           int i, int j, int W) {
    float v = im[i * W + j] - p0[i * W + j] - p1[i * W + j];
    if (i > 0) v += p0[(i - 1) * W + j];
    if (j > 0) v += p1[i * W + j - 1];
    return v;
}

__global__ void k_tv_step(const float* __restrict__ img, const float* __restrict__ pin,
                          float* __restrict__ pout, int H, int W, int total) {
    int idx = blockIdx.x * blockDim.x + threadIdx.x;
    if (idx >= total) return;
    int j = idx % W;
    int i = (idx / W) % H;
    int n = idx / (W * H);
    const float* im = img + (size_t)n * H * W;
    const float* p0 = pin + (size_t)n * 2 * H * W;
    const float* p1 = p0 + H * W;
    float o00 = tv_outval(im, p0, p1, i, j, W);
    float gy = 0.0f, gx = 0.0f;
    if (i < H - 1) gy = tv_outval(im, p0, p1, i + 1, j, W) - o00;
    if (j < W - 1) gx = tv_outval(im, p0, p1, i, j + 1, W) - o00;
    float nrm = sqrtf(gy * gy + gx * gx);
    float inv = 1.0f / (1.0f + TV_RATIO * nrm);
    float* q0 = pout + (size_t)n * 2 * H * W;
    float* q1 = q0 + H * W;
    q0[i * W + j] = (p0[i * W + j] - TV_TAU * gy) * inv;
    q1[i * W + j] = (p1[i * W + j] - TV_TAU * gx) * inv;
}

__global__ void k_tv_out(const float* __restrict__ img, const float* __restrict__ pin,
                         float* __restrict__ den, int H, int W, int total) {
    int idx = blockIdx.x * blockDim.x + threadIdx.x;
    if (idx >= total) return;
    int j = idx % W;
    int i = (idx / W) % H;
    int n = idx / (W * H);
    const float* im = img + (size_t)n * H * W;
    const float* p0 = pin + (size_t)n * 2 * H * W;
    const float* p1 = p0 + H * W;
    den[idx] = tv_outval(im, p0, p1, i, j, W);
}

// ---------------------------------------------------------------- stem 3x3 s2, 1->32, SiLU
__global__ void k_stem(const float* __restrict__ in, const float* __restrict__ w,
                       float* __restrict__ out, int Hin, int Win, int Hout, int Wout,
                       int total) {
    int idx = blockIdx.x * blockDim.x + threadIdx.x;
    if (idx >= total) return;
    int x = idx % Wout;
    int y = (idx / Wout) % Hout;
    int o = (idx / (Wout * Hout)) % 32;
    int b = idx / (Wout * Hout * 32);
    const float* ib = in + (size_t)b * Hin * Win;
    const float* wo = w + o * 9;
    float s = 0.0f;
    for (int ky = 0; ky < 3; ++ky) {
        int iy = 2 * y + ky - 1;
        if (iy < 0 || iy >= Hin) continue;
        for (int kx = 0; kx < 3; ++kx) {
            int ix = 2 * x + kx - 1;
            if (ix < 0 || ix >= Win) continue;
            s += wo[ky * 3 + kx] * ib[iy * Win + ix];
        }
    }
    out[idx] = silu_f(s);
}

// ---------------------------------------------------------------- depthwise 3x3 s2, SiLU
__global__ void k_dw(const float* __restrict__ in, const float* __restrict__ w,
                     float* __restrict__ out, int C, int Hin, int Win, int Hout, int Wout,
                     int total) {
    int idx = blockIdx.x * blockDim.x + threadIdx.x;
    if (idx >= total) return;
    int x = idx % Wout;
    int y = (idx / Wout) % Hout;
    int c = (idx / (Wout * Hout)) % C;
    int b = idx / (Wout * Hout * C);
    const float* ib = in + ((size_t)b * C + c) * Hin * Win;
    const float* wc = w + c * 9;
    float s = 0.0f;
    for (int ky = 0; ky < 3; ++ky) {
        int iy = 2 * y + ky - 1;
        if (iy < 0 || iy >= Hin) continue;
        for (int kx = 0; kx < 3; ++kx) {
            int ix = 2 * x + kx - 1;
            if (ix < 0 || ix >= Win) continue;
            s += wc[ky * 3 + kx] * ib[iy * Win + ix];
        }
    }
    out[idx] = silu_f(s);
}

// ---------------------------------------------------------------- 1x1 conv as GEMM via fp32 WMMA
// out[(b*M+m)*HW + hw] = act( sum_k W[m*K+k] * in[(b*K+k)*HW + hw] + bias[m] )
// One wave computes a 16(M) x 32(N) tile: two D accumulators sharing one A fetch
// per K-step of 4, using V_WMMA_F32_16X16X4_F32 (exact fp32 semantics).
//   A 16x4 (MxK): lanes 0-15 -> M=lane, K={k,k+1}; lanes 16-31 -> M=lane-16, K={k+2,k+3}
//   B 4x16 (KxN): lanes 0-15 -> N=lane, K={k,k+1}; lanes 16-31 -> N=lane-16, K={k+2,k+3}
//   D 16x16: VGPR r -> M = r (lanes 0-15) / r+8 (lanes 16-31), N = lane&15
__global__ void k_pw_wmma(const float* __restrict__ in, const float* __restrict__ W,
                          const float* __restrict__ bias, float* __restrict__ out,
                          int K, int M, int HW, int act) {
    int lane = threadIdx.x;          // block = 32 threads = one wave32
    int n0 = blockIdx.x * 32;        // position along N = B*HW (HW % 32 == 0)
    int m0 = blockIdx.y * 16;        // position along M (output channels)
    int b  = n0 / HW;
    int hw = n0 - b * HW;
    int l    = lane & 15;
    int hi   = lane >> 4;            // 0 for lanes 0-15, 1 for 16-31
    int koff = hi << 1;              // K sub-offset 0 or 2

    const float* a_ptr = W + (size_t)(m0 + l) * K + koff;
    const float* b_ptr = in + ((size_t)b * K + koff) * HW + hw + l;
    const size_t bstep = (size_t)4 * HW;

    v8f acc0 = {};
    v8f acc1 = {};
    for (int k = 0; k < K; k += 4) {
        v2f a, b0, b1;
        a.x  = a_ptr[0];
        a.y  = a_ptr[1];
        b0.x = b_ptr[0];
        b0.y = b_ptr[HW];
        b1.x = b_ptr[16];
        b1.y = b_ptr[HW + 16];
        acc0 = __builtin_amdgcn_wmma_f32_16x16x4_f32(
            false, a, false, b0, (short)0, acc0, false, false);
        acc1 = __builtin_amdgcn_wmma_f32_16x16x4_f32(
            false, a, false, b1, (short)0, acc1, false, false);
        a_ptr += 4;
        b_ptr += bstep;
    }

    int mb = m0 + (hi << 3);         // +8 for the high half-wave
    float* ob = out + (size_t)b * M * HW + hw + l;
    for (int r = 0; r < 8; ++r) {
        int m = mb + r;
        float bv = bias ? bias[m] : 0.0f;
        float v0 = acc0[r] + bv;
        float v1 = acc1[r] + bv;
        if (act) { v0 = silu_f(v0); v1 = silu_f(v1); }
        float* o = ob + (size_t)m * HW;
        o[0]  = v0;
        o[16] = v1;
    }
}

// ---------------------------------------------------------------- fused bilinear 16->512 + 2-level Haar LL
// in: [B*64, 16, 16]  ->  out: [B*64, 128, 128]; out = 0.25 * sum over 4x4 of upsampled.
__global__ void k_resize_haar(const float* __restrict__ in, float* __restrict__ out, int total) {
    int idx = blockIdx.x * blockDim.x + threadIdx.x;
    if (idx >= total) return;
    int X  = idx & 127;
    int Y  = (idx >> 7) & 127;
    int bc = idx >> 14;
    const float* p = in + (size_t)bc * 256;
    float s = 0.0f;
    for (int sy = 0; sy < 4; ++sy) {
        int y = 4 * Y + sy;
        float fy = (y + 0.5f) * 0.03125f - 0.5f;   // scale 16/512
        int y0 = (int)floorf(fy);
        float wy = fy - (float)y0;
        int y0c = min(max(y0, 0), 15);
        int y1c = min(max(y0 + 1, 0), 15);
        for (int sx = 0; sx < 4; ++sx) {
            int x = 4 * X + sx;
            float fx = (x + 0.5f) * 0.03125f - 0.5f;
            int x0 = (int)floorf(fx);
            float wx = fx - (float)x0;
            int x0c = min(max(x0, 0), 15);
            int x1c = min(max(x0 + 1, 0), 15);
            float v00 = p[y0c * 16 + x0c], v01 = p[y0c * 16 + x1c];
            float v10 = p[y1c * 16 + x0c], v11 = p[y1c * 16 + x1c];
            float top = v00 + wx * (v01 - v00);
            float bot = v10 + wx * (v11 - v10);
            s += top + wy * (bot - top);
        }
    }
    out[idx] = 0.25f * s;
}

// ---------------------------------------------------------------- final 1x1 conv 64->1 + bias
__global__ void k_final(const float* __restrict__ ll, const float* __restrict__ w,
                        const float* __restrict__ bias, float* __restrict__ out, int total) {
    int idx = blockIdx.x * blockDim.x + threadIdx.x;
    if (idx >= total) return;
    int pix = idx & 16383;           // 128*128
    int b = idx >> 14;
    const float* base = ll + (size_t)b * 64 * 16384 + pix;
    float s = bias[0];
    for (int c = 0; c < 64; ++c) s += w[c] * base[(size_t)c * 16384];
    out[idx] = s;
}

// ================================================================ host side
extern "C" void kernel_launch(void* const* d_in, const int* in_sizes, int n_in,
                              void* d_out, int out_size, void* d_ws, size_t ws_size,
                              hipStream_t stream) {
    (void)in_sizes; (void)n_in; (void)out_size; (void)ws_size;
    const float* img     = (const float*)d_in[0];
    const float* stem_w  = (const float*)d_in[1];
    const float* dw1_w   = (const float*)d_in[2];
    const float* pw1_w   = (const float*)d_in[3];
    const float* dw2_w   = (const float*)d_in[4];
    const float* pw2_w   = (const float*)d_in[5];
    const float* dw3_w   = (const float*)d_in[6];
    const float* pw3_w   = (const float*)d_in[7];
    const float* dw4_w   = (const float*)d_in[8];
    const float* pw4_w   = (const float*)d_in[9];
    const float* head_w  = (const float*)d_in[10];
    const float* proj_w  = (const float*)d_in[11];
    const float* proj_b  = (const float*)d_in[12];
    const float* final_w = (const float*)d_in[13];
    const float* final_b = (const float*)d_in[14];
    float* outp = (float*)d_out;

    char* ws = (char*)d_ws;
    float* pA  = (float*)(ws);                         // 8 MB  : p ping
    float* pB  = (float*)(ws + (size_t)8  * 1048576);  // 8 MB  : p pong / denoised
    float* X   = (float*)(ws + (size_t)16 * 1048576);  // 32 MB : conv ping
    float* Y   = (float*)(ws + (size_t)48 * 1048576);  // 16 MB : conv pong

    const int Bn = 4, H = 512, Wd = 512;
    const int npix = Bn * H * Wd;                      // 1,048,576

    // ---- TV-Chambolle: zero p, 20 steps, emit denoised into pB
    k_zero<<<(2 * npix + 255) / 256, 256, 0, stream>>>(pA, 2 * npix);
    float* pin = pA; float* pout = pB;
    for (int it = 0; it < 20; ++it) {
        k_tv_step<<<npix / 256, 256, 0, stream>>>(img, pin, pout, H, Wd, npix);
        float* t = pin; pin = pout; pout = t;
    }
    // after 20 swaps pin == pA; denoised goes to pB (p data there is dead)
    float* den = pB;
    k_tv_out<<<npix / 256, 256, 0, stream>>>(img, pin, den, H, Wd, npix);

    // ---- stem: [4,1,512,512] -> [4,32,256,256]  (X)
    {
        int total = Bn * 32 * 256 * 256;
        k_stem<<<total / 256, 256, 0, stream>>>(den, stem_w, X, 512, 512, 256, 256, total);
    }
    // ---- stage 1: dw 32ch 256->128 (Y), pw 32->64 (X)
    {
        int total = Bn * 32 * 128 * 128;
        k_dw<<<total / 256, 256, 0, stream>>>(X, dw1_w, Y, 32, 256, 256, 128, 128, total);
        int HW = 128 * 128, N = Bn * HW;
        k_pw_wmma<<<dim3(N / 32, 64 / 16), 32, 0, stream>>>(Y, pw1_w, nullptr, X, 32, 64, HW, 1);
    }
    // ---- stage 2: dw 64ch 128->64 (Y), pw 64->128 (X)
    {
        int total = Bn * 64 * 64 * 64;
        k_dw<<<total / 256, 256, 0, stream>>>(X, dw2_w, Y, 64, 128, 128, 64, 64, total);
        int HW = 64 * 64, N = Bn * HW;
        k_pw_wmma<<<dim3(N / 32, 128 / 16), 32, 0, stream>>>(Y, pw2_w, nullptr, X, 64, 128, HW, 1);
    }
    // ---- stage 3: dw 128ch 64->32 (Y), pw 128->256 (X)
    {
        int total = Bn * 128 * 32 * 32;
        k_dw<<<total / 256, 256, 0, stream>>>(X, dw3_w, Y, 128, 64, 64, 32, 32, total);
        int HW = 32 * 32, N = Bn * HW;
        k_pw_wmma<<<dim3(N / 32, 256 / 16), 32, 0, stream>>>(Y, pw3_w, nullptr, X, 128, 256, HW, 1);
    }
    // ---- stage 4: dw 256ch 32->16 (Y), pw 256->512 (X)
    {
        int total = Bn * 256 * 16 * 16;
        k_dw<<<total / 256, 256, 0, stream>>>(X, dw4_w, Y, 256, 32, 32, 16, 16, total);
        int HW = 16 * 16, N = Bn * HW;
        k_pw_wmma<<<dim3(N / 32, 512 / 16), 32, 0, stream>>>(Y, pw4_w, nullptr, X, 256, 512, HW, 1);
    }
    // ---- head 512->1280 + SiLU (X -> Y), proj 1280->64 + bias (Y -> X)
    {
        int HW = 16 * 16, N = Bn * HW;
        k_pw_wmma<<<dim3(N / 32, 1280 / 16), 32, 0, stream>>>(X, head_w, nullptr, Y, 512, 1280, HW, 1);
        k_pw_wmma<<<dim3(N / 32, 64 / 16), 32, 0, stream>>>(Y, proj_w, proj_b, X, 1280, 64, HW, 0);
    }
    // ---- fused bilinear upsample (16->512) + 2-level Haar LL: X -> Y [4,64,128,128]
    {
        int total = Bn * 64 * 128 * 128;
        k_resize_haar<<<total / 256, 256, 0, stream>>>(X, Y, total);
    }
    // ---- final 1x1 conv 64->1 + bias -> d_out [4,1,128,128]
    {
        int total = Bn * 128 * 128;
        k_final<<<total / 256, 256, 0, stream>>>(Y, final_w, final_b, outp, total);
    }
}